// ModelClass_1133871366191
// MI455X (gfx1250) — compile-verified
//
#include <hip/hip_runtime.h>

#define E   128
#define NF  64
#define NEG 0.2f

typedef __attribute__((ext_vector_type(2))) float v2f;
typedef __attribute__((ext_vector_type(8))) float v8f;

__device__ __forceinline__ float leaky_f(float v) { return v > 0.f ? v : NEG * v; }

// ---------------------------------------------------------------------------
// Weight pre-pack: Wp[(k/2)*2*N + n*2 + (k&1)] = W[k*N + n]
// Makes each B fragment (W[k][n], W[k+1][n]) one 8-byte load in the GEMM.
// ---------------------------------------------------------------------------
__global__ void pack_w_kernel(const float* __restrict__ W, float* __restrict__ Wp,
                              int Ncols, long total) {
  long idx = (long)blockIdx.x * blockDim.x + threadIdx.x;
  if (idx >= total) return;
  long k = idx / Ncols;
  int n = (int)(idx % Ncols);
  Wp[(size_t)(k >> 1) * 2 * Ncols + n * 2 + (k & 1)] = W[idx];
}

// ---------------------------------------------------------------------------
// WMMA fp32 GEMM: Y[M, NT*16] = leaky( (X [+ Xadd]) [M,K] @ W[K,N] + bias[N] )
// V_WMMA_F32_16X16X4_F32; wave owns MT*16 rows x N cols (MT*NT 16x16 tiles).
// A frag lane L: rows M = r0 + mi*16 + (L&15), K pair {k0,k0+1} (lanes 0-15)
// or {k0+2,k0+3} (lanes 16-31). B frag from packed Wp: one b64 per tile.
// C/D: VGPR j = (M = j + 8*(L>=16), N = L&15).
// ---------------------------------------------------------------------------
template <int NT, int MT, bool HAS_ADD>
__global__ __launch_bounds__(256) void mlp_wmma(const float* __restrict__ X,
                                                const float* __restrict__ Xadd,
                                                const float* __restrict__ Wp,
                                                const float* __restrict__ bias,
                                                float* __restrict__ Y,
                                                int M, int K) {
  constexpr int N = NT * 16;
  const int lane = threadIdx.x & 31;
  const int wave = threadIdx.x >> 5;
  const int r0 = blockIdx.x * (128 * MT) + wave * (16 * MT);
  if (r0 >= M) return;  // wave-uniform exit: EXEC stays all-ones in WMMA waves
  const int lo   = lane & 15;
  const int hi   = lane >> 4;
  const int koff = hi * 2;

  v8f acc[MT][NT];
#pragma unroll
  for (int t = 0; t < NT; ++t) {
    float bv = bias[t * 16 + lo];
#pragma unroll
    for (int mi = 0; mi < MT; ++mi)
#pragma unroll
      for (int j = 0; j < 8; ++j) acc[mi][t][j] = bv;
  }

  const float* xrow[MT];
  const float* arow[MT];
#pragma unroll
  for (int mi = 0; mi < MT; ++mi) {
    size_t m = (size_t)(r0 + mi * 16 + lo);
    xrow[mi] = X + m * K;
    if (HAS_ADD) arow[mi] = Xadd + m * K;
  }

  for (int k0 = 0; k0 < K; k0 += 4) {
    v2f a[MT];
#pragma unroll
    for (int mi = 0; mi < MT; ++mi) {
      a[mi].x = xrow[mi][k0 + koff];
      a[mi].y = xrow[mi][k0 + koff + 1];
      if (HAS_ADD) {
        a[mi].x += arow[mi][k0 + koff];
        a[mi].y += arow[mi][k0 + koff + 1];
      }
    }
#pragma unroll
    for (int t = 0; t < NT; ++t) {
      v2f b = *(const v2f*)(Wp + (size_t)(k0 + koff) * N + (t * 16 + lo) * 2);
#pragma unroll
      for (int mi = 0; mi < MT; ++mi) {
        acc[mi][t] = __builtin_amdgcn_wmma_f32_16x16x4_f32(
            false, a[mi], false, b, (short)0, acc[mi][t], false, false);
      }
    }
  }

#pragma unroll
  for (int mi = 0; mi < MT; ++mi)
#pragma unroll
    for (int t = 0; t < NT; ++t)
#pragma unroll
      for (int j = 0; j < 8; ++j) {
        int row = r0 + mi * 16 + hi * 8 + j;
        Y[(size_t)row * N + t * 16 + lo] = leaky_f(acc[mi][t][j]);
      }
}

// ---------------------------------------------------------------------------
// Segment mean/max over event = row % E; writes concat [mean | max] (E x 128)
// H is (cnt*E, 64) -> float4 lanes; one block per event, 16 partials/feature4.
// ---------------------------------------------------------------------------
__global__ __launch_bounds__(256) void seg_reduce_kernel(const float* __restrict__ H,
                                                         float* __restrict__ GCAT,
                                                         int cnt) {
  __shared__ float4 ssum[256];
  __shared__ float4 smax[256];
  const int e  = blockIdx.x;
  const int f4 = threadIdx.x & 15;  // features f4*4 .. f4*4+3
  const int p  = threadIdx.x >> 4;  // 16 partials
  const float ninf = -__builtin_inff();
  float4 s  = {0.f, 0.f, 0.f, 0.f};
  float4 mx = {ninf, ninf, ninf, ninf};
  const float4* H4 = (const float4*)H;
  for (int n = p; n < cnt; n += 16) {
    float4 v = H4[((size_t)n * E + e) * 16 + f4];
    s.x += v.x; s.y += v.y; s.z += v.z; s.w += v.w;
    mx.x = fmaxf(mx.x, v.x); mx.y = fmaxf(mx.y, v.y);
    mx.z = fmaxf(mx.z, v.z); mx.w = fmaxf(mx.w, v.w);
  }
  ssum[threadIdx.x] = s;
  smax[threadIdx.x] = mx;
  __syncthreads();
  if (threadIdx.x < 16) {
    float4 S  = {0.f, 0.f, 0.f, 0.f};
    float4 Mx = {ninf, ninf, ninf, ninf};
    for (int j = 0; j < 16; ++j) {
      float4 a = ssum[j * 16 + f4];
      float4 b = smax[j * 16 + f4];
      S.x += a.x; S.y += a.y; S.z += a.z; S.w += a.w;
      Mx.x = fmaxf(Mx.x, b.x); Mx.y = fmaxf(Mx.y, b.y);
      Mx.z = fmaxf(Mx.z, b.z); Mx.w = fmaxf(Mx.w, b.w);
    }
    float inv = 1.f / (float)cnt;
    float4* G4 = (float4*)GCAT;
    float4 Sm = {S.x * inv, S.y * inv, S.z * inv, S.w * inv};
    G4[(size_t)e * 32 + f4]      = Sm;
    G4[(size_t)e * 32 + 16 + f4] = Mx;
  }
}

// OUT[i,0:64] = Xs[i], OUT[i,64:128] = glob[i % E]  -- float4 lanes, 32/row
__global__ void cat_glob_kernel(const float* __restrict__ Xs, const float* __restrict__ glob,
                                float* __restrict__ OUT, long total4) {
  long idx = (long)blockIdx.x * blockDim.x + threadIdx.x;
  if (idx >= total4) return;
  long row = idx >> 5;
  int f4 = (int)(idx & 31);
  const float4* X4 = (const float4*)Xs;
  const float4* G4 = (const float4*)glob;
  ((float4*)OUT)[idx] = (f4 < 16) ? X4[row * 16 + f4] : G4[(row & 127) * 16 + (f4 - 16)];
}

// ch (Ml x 128) -> children: row (p,e), col chunk c -> xApp[(p*2+c)*E + e, :]
__global__ void scatter_children_kernel(const float* __restrict__ CH,
                                        float* __restrict__ XAPP, long total4) {
  long idx = (long)blockIdx.x * blockDim.x + threadIdx.x;
  if (idx >= total4) return;
  long i = idx >> 5;
  int f4 = (int)(idx & 31);
  long p = i >> 7;
  int e = (int)(i & 127);
  int c = f4 >> 4;
  int g4 = f4 & 15;
  long orow = (p * 2 + c) * 128 + e;
  ((float4*)XAPP)[orow * 16 + g4] = ((const float4*)CH)[idx];
}

// anc[level l] = anc[parent] + hx[parent]  (one launch per level, float4)
__global__ void anc_level_kernel(float* __restrict__ ANC, const float* __restrict__ HX,
                                 long rowBase, long parentBase, long total4) {
  long idx = (long)blockIdx.x * blockDim.x + threadIdx.x;
  if (idx >= total4) return;
  long i = idx >> 5;
  int f4 = (int)(idx & 31);
  long k = i >> 7;
  int e = (int)(i & 127);
  long row  = rowBase + i;
  long prow = parentBase + (k >> 1) * 128 + e;
  float4 a = ((const float4*)ANC)[prow * 32 + f4];
  float4 h = ((const float4*)HX)[prow * 32 + f4];
  float4 r = {a.x + h.x, a.y + h.y, a.z + h.z, a.w + h.w};
  ((float4*)ANC)[row * 32 + f4] = r;
}

__global__ void fill_zero_kernel(float* __restrict__ p, long n4) {
  long i = (long)blockIdx.x * blockDim.x + threadIdx.x;
  if (i < n4) ((float4*)p)[i] = (float4){0.f, 0.f, 0.f, 0.f};
}

__global__ void copy_kernel(const float* __restrict__ a, float* __restrict__ b, long n) {
  long i = (long)blockIdx.x * blockDim.x + threadIdx.x;
  if (i < n) b[i] = a[i];
}

__global__ void out_kernel(const float* __restrict__ x, float* __restrict__ out, long rows) {
  long idx = (long)blockIdx.x * blockDim.x + threadIdx.x;
  if (idx >= rows * 3) return;
  long i = idx / 3;
  int j = (int)(idx % 3);
  out[idx] = x[i * 64 + j];
}

// ---------------------------------------------------------------------------

static inline dim3 ew(long n) { return dim3((unsigned)((n + 255) / 256)); }

static inline void launch_mlp(const float* X, const float* Xadd, const float* Wp,
                              const float* b, float* Y, long M, int K, int N,
                              hipStream_t s) {
  dim3 grid((unsigned)((M + 255) / 256)), block(256);
  if (N == 64) {
    if (Xadd) mlp_wmma<4, 2, true><<<grid, block, 0, s>>>(X, Xadd, Wp, b, Y, (int)M, K);
    else      mlp_wmma<4, 2, false><<<grid, block, 0, s>>>(X, nullptr, Wp, b, Y, (int)M, K);
  } else {
    if (Xadd) mlp_wmma<8, 2, true><<<grid, block, 0, s>>>(X, Xadd, Wp, b, Y, (int)M, K);
    else      mlp_wmma<8, 2, false><<<grid, block, 0, s>>>(X, nullptr, Wp, b, Y, (int)M, K);
  }
}

extern "C" void kernel_launch(void* const* d_in, const int* in_sizes, int n_in,
                              void* d_out, int out_size, void* d_ws, size_t ws_size,
                              hipStream_t stream) {
  const float* rv      = (const float*)d_in[0];
  const float* pre_W   = (const float*)d_in[1];
  const float* pre_b   = (const float*)d_in[2];
  const float* post_W0 = (const float*)d_in[3];
  const float* post_b0 = (const float*)d_in[4];
  const float* post_W1 = (const float*)d_in[5];
  const float* post_b1 = (const float*)d_in[6];
  const float* post_Wr = (const float*)d_in[7];
  const float* post_br = (const float*)d_in[8];
  const float* proj_W  = (const float*)d_in[9];
  const float* proj_b  = (const float*)d_in[10];
  const float* gin_W0  = (const float*)d_in[11];
  const float* gin_b0  = (const float*)d_in[12];
  const float* gin_W1  = (const float*)d_in[13];
  const float* gin_b1  = (const float*)d_in[14];
  const float* gin_Wr  = (const float*)d_in[15];
  const float* gin_br  = (const float*)d_in[16];

  static const long offs[11] = {0, 1, 3, 7, 15, 31, 63, 127, 255, 511, 1023};
  const long MAXN = 1023L * 128;  // 130944 rows

  float* ws   = (float*)d_ws;
  float* x    = ws;                   // MAXN x 64
  float* T0   = x + MAXN * 64;        // MAXN x 128
  float* T1   = T0 + MAXN * 128;      // MAXN x 128
  float* HX   = T1 + MAXN * 128;      // MAXN x 128
  float* ANC  = HX + MAXN * 128;      // MAXN x 128
  float* GCAT = ANC + MAXN * 128;     // 128 x 128
  float* GT0  = GCAT + 128 * 128;     // 128 x 128
  float* GT1  = GT0 + 128 * 128;      // 128 x 128
  // packed weights
  float* p_pre  = GT1 + 128 * 128;            // 5*4096
  float* p_pW0  = p_pre + 5 * 4096;           // 16384
  float* p_pW1  = p_pW0 + 16384;              // 8192
  float* p_pWr  = p_pW1 + 8192;               // 3*4096
  float* p_proj = p_pWr + 3 * 4096;           // 5*16384
  float* p_gW0  = p_proj + 5 * 16384;         // 16384
  float* p_gW1  = p_gW0 + 16384;              // 8192
  float* p_gWr  = p_gW1 + 8192;               // 3*4096

  // ---- pre-pack all weight matrices (tiny, once per launch)
  auto packW = [&](const float* W, float* Wp, int K, int N) {
    long tot = (long)K * N;
    pack_w_kernel<<<ew(tot), 256, 0, stream>>>(W, Wp, N, tot);
  };
  for (int i = 0; i < 5; ++i) packW(pre_W + (size_t)i * 4096, p_pre + (size_t)i * 4096, 64, 64);
  packW(post_W0, p_pW0, 128, 128);
  packW(post_W1, p_pW1, 128, 64);
  for (int i = 0; i < 3; ++i) packW(post_Wr + (size_t)i * 4096, p_pWr + (size_t)i * 4096, 64, 64);
  for (int i = 0; i < 5; ++i) packW(proj_W + (size_t)i * 16384, p_proj + (size_t)i * 16384, 128, 128);
  packW(gin_W0, p_gW0, 128, 128);
  packW(gin_W1, p_gW1, 128, 64);
  for (int i = 0; i < 3; ++i) packW(gin_Wr + (size_t)i * 4096, p_gWr + (size_t)i * 4096, 64, 64);

  copy_kernel<<<ew(128 * 64), 256, 0, stream>>>(rv, x, 128 * 64);

  int n_lvl = 1;
  for (int s = 0; s < 11; ++s) {  // L-1 + POST_STEPS = 11 steps
    long N = offs[n_lvl] * E;

    // ---- pre MLP: 5 x (64 -> 64), ping-pong x -> T0 -> T1 -> ... -> T0
    {
      const float* cur = x;
      float* bufs[2] = {T0, T1};
      for (int i = 0; i < 5; ++i) {
        float* nxt = bufs[i & 1];
        launch_mlp(cur, nullptr, p_pre + (size_t)i * 4096, pre_b + i * 64,
                   nxt, N, 64, 64, stream);
        cur = nxt;
      }
      // h lands in T0
    }

    // ---- segment mean/max -> GCAT (E x 128, [mean|max])
    seg_reduce_kernel<<<128, 256, 0, stream>>>(T0, GCAT, (int)offs[n_lvl]);

    // ---- post MLP (mix): 128->128, 128->64, 3 x 64->64   (glob ends in GT0)
    launch_mlp(GCAT, nullptr, p_pW0, post_b0, GT0, 128, 128, 128, stream);
    launch_mlp(GT0, nullptr, p_pW1, post_b1, GT1, 128, 128, 64, stream);
    launch_mlp(GT1, nullptr, p_pWr + 0 * 4096, post_br + 0 * 64, GT0, 128, 64, 64, stream);
    launch_mlp(GT0, nullptr, p_pWr + 1 * 4096, post_br + 1 * 64, GT1, 128, 64, 64, stream);
    launch_mlp(GT1, nullptr, p_pWr + 2 * 4096, post_br + 2 * 64, GT0, 128, 64, 64, stream);
    float* glob = GT0;

    // ---- growth: project last level + scatter children
    if (s < 9) {
      int lvl = n_lvl - 1;
      long start = offs[lvl] * E;
      long Ml = (offs[n_lvl] - offs[lvl]) * E;  // B^lvl * E rows
      cat_glob_kernel<<<ew(Ml * 32), 256, 0, stream>>>(x + start * 64, glob, T1, Ml * 32);
      {
        const float* cur = T1;
        float* bufs[2] = {T0, T1};
        for (int i = 0; i < 5; ++i) {
          float* nxt = bufs[i & 1];
          launch_mlp(cur, nullptr, p_proj + (size_t)i * 16384, proj_b + i * 128,
                     nxt, Ml, 128, 128, stream);
          cur = nxt;
        }
        // ch lands in T0
      }
      scatter_children_kernel<<<ew(Ml * 32), 256, 0, stream>>>(T0, x + N * 64, Ml * 32);
      n_lvl++;
      N = offs[n_lvl] * E;
    }

    // ---- hx = [x | glob[event]]
    cat_glob_kernel<<<ew(N * 32), 256, 0, stream>>>(x, glob, HX, N * 32);

    // ---- ancestor sums, level by level (replaces edge segment_sum)
    fill_zero_kernel<<<ew((long)E * 32), 256, 0, stream>>>(ANC, (long)E * 32);
    for (int l = 1; l < n_lvl; ++l) {
      long cntl = (offs[l + 1] - offs[l]) * E;  // B^l * E rows
      anc_level_kernel<<<ew(cntl * 32), 256, 0, stream>>>(
          ANC, HX, offs[l] * E, offs[l - 1] * E, cntl * 32);
    }

    // ---- gin MLP on (hx + anc): 128->128, 128->64, 3 x 64->64, final -> x
    launch_mlp(HX, ANC, p_gW0, gin_b0, T0, N, 128, 128, stream);
    launch_mlp(T0, nullptr, p_gW1, gin_b1, T1, N, 128, 64, stream);
    launch_mlp(T1, nullptr, p_gWr + 0 * 4096, gin_br + 0 * 64, T0, N, 64, 64, stream);
    launch_mlp(T0, nullptr, p_gWr + 1 * 4096, gin_br + 1 * 64, T1, N, 64, 64, stream);
    launch_mlp(T1, nullptr, p_gWr + 2 * 4096, gin_br + 2 * 64, x, N, 64, 64, stream);
  }

  out_kernel<<<ew(MAXN * 3), 256, 0, stream>>>(x, (float*)d_out, MAXN);
}